// DistMultDecoder_12232066859179
// MI455X (gfx1250) — compile-verified
//
#include <hip/hip_runtime.h>

typedef __attribute__((ext_vector_type(2))) float v2f;
typedef __attribute__((ext_vector_type(8))) float v8f;

#define TILE    128
#define KC      32
#define KSTRIDE 36                     // padded LDS row stride (floats): 16B-aligned, bank-conflict-free
#define D_DIM   512
#define NCHUNK  (D_DIM / KC)           // 16
#define BUFBYTES (TILE * KSTRIDE * 4)  // 18432 bytes per tile buffer

// CDNA5 async memory->LDS copy (no VGPR transit, tracked by ASYNCcnt).
// GVS form: mem_addr = SGPR64 + VGPR32 + IOFFSET ; vdst = LDS byte address.
__device__ __forceinline__ void async_copy_b128(unsigned ldsByteAddr,
                                                unsigned long long gbase,
                                                unsigned gByteOff) {
    asm volatile("global_load_async_to_lds_b128 %0, %1, %2"
                 :: "v"(ldsByteAddr), "v"(gByteOff), "s"(gbase)
                 : "memory");
}
__device__ __forceinline__ void wait_async_le8() {
    asm volatile("s_wait_asynccnt 0x8" ::: "memory");
}
__device__ __forceinline__ void wait_async_0() {
    asm volatile("s_wait_asynccnt 0x0" ::: "memory");
}

// Issue one K-chunk's A/B tiles (128 rows x 32 K floats each) as 8 async b128 ops/thread-iter.
__device__ __forceinline__ void stage_async(int tid, int rowBase, int colBase, int kc0,
                                            unsigned long long gbase,
                                            unsigned ldsA, unsigned ldsB) {
#pragma unroll
    for (int i = 0; i < 4; ++i) {
        const int idx = tid + i * 256;            // 0..1023 float4 slots
        const int row = idx >> 3;                 // 0..127
        const unsigned kv = (idx & 7) << 2;       // 0,4,...,28
        const unsigned lofs = (unsigned)(row * KSTRIDE + kv) * 4u;
        const unsigned gA = ((unsigned)(rowBase + row) * D_DIM + kc0 + kv) * 4u;
        const unsigned gB = ((unsigned)(colBase + row) * D_DIM + kc0 + kv) * 4u;
        async_copy_b128(ldsA + lofs, gbase, gA);
        async_copy_b128(ldsB + lofs, gbase, gB);
    }
}

__global__ __launch_bounds__(256)
void distmult_wmma_f32_kernel(const float* __restrict__ inputs,
                              const float* __restrict__ relation,
                              float* __restrict__ out, int N) {
    __shared__ float sAb[2][TILE * KSTRIDE];   // double-buffered raw row block
    __shared__ float sBb[2][TILE * KSTRIDE];   // double-buffered raw column block
    __shared__ float sRel[D_DIM];              // relation, staged once

    const int tid  = threadIdx.x;
    const int lane = tid & 31;
    const int wave = tid >> 5;
    const int wm   = wave & 3;     // 4 row slabs of 32
    const int wn   = wave >> 2;    // 2 col slabs of 64

    const int rowBase = blockIdx.y * TILE;
    const int colBase = blockIdx.x * TILE;

    const int half = lane >> 4;    // lane half selects K pair (ISA 7.12.2, 32-bit A/B)
    const int l15  = lane & 15;

    const unsigned long long gbase = (unsigned long long)inputs;
    // LDS byte offsets (flat-aperture low 32 bits == wave-relative DS address)
    const unsigned ldsAbase = (unsigned)(size_t)&sAb[0][0];
    const unsigned ldsBbase = (unsigned)(size_t)&sBb[0][0];

    v8f acc[2][4];
#pragma unroll
    for (int i = 0; i < 2; ++i)
#pragma unroll
        for (int j = 0; j < 4; ++j)
            acc[i][j] = (v8f){0.f,0.f,0.f,0.f,0.f,0.f,0.f,0.f};

    // Prologue: async-stage chunk 0 into buffer 0; stage relation into LDS.
    stage_async(tid, rowBase, colBase, 0, gbase, ldsAbase, ldsBbase);
    *(float2*)&sRel[tid * 2] = *(const float2*)&relation[tid * 2];

#pragma unroll 1
    for (int c = 0; c < NCHUNK; ++c) {
        const int p = c & 1;
        const int kc0 = c * KC;

        if (c + 1 < NCHUNK) {
            // Kick off next chunk into the other buffer, then wait so that only
            // the 8 new ops remain in flight (async loads complete in order).
            stage_async(tid, rowBase, colBase, kc0 + KC, gbase,
                        ldsAbase + (unsigned)(1 - p) * BUFBYTES,
                        ldsBbase + (unsigned)(1 - p) * BUFBYTES);
            wait_async_le8();
        } else {
            wait_async_0();
        }
        __syncthreads();   // chunk-c tiles (+ sRel on c==0) visible to all waves

        const float* __restrict__ tA = sAb[p];
        const float* __restrict__ tB = sBb[p];

        // ---- 8 k-steps of V_WMMA_F32_16X16X4_F32 over this chunk ----
#pragma unroll
        for (int ks = 0; ks < 8; ++ks) {
            const int kOff = (ks << 2) + (half << 1);   // consecutive K pair per lane
            const v2f r = *(const v2f*)(&sRel[kc0 + kOff]);
            v2f a[2], b[4];
#pragma unroll
            for (int tm = 0; tm < 2; ++tm) {
                a[tm] = *(const v2f*)(&tA[(wm * 32 + tm * 16 + l15) * KSTRIDE + kOff]);
                a[tm] = a[tm] * r;   // fused diag(relation): matches reference rounding
            }
#pragma unroll
            for (int tn = 0; tn < 4; ++tn)
                b[tn] = *(const v2f*)(&tB[(wn * 64 + tn * 16 + l15) * KSTRIDE + kOff]);
#pragma unroll
            for (int tm = 0; tm < 2; ++tm)
#pragma unroll
                for (int tn = 0; tn < 4; ++tn)
                    acc[tm][tn] = __builtin_amdgcn_wmma_f32_16x16x4_f32(
                        false, a[tm], false, b[tn],
                        (short)0, acc[tm][tn], false, false);
        }
        __syncthreads();   // all waves done reading buffer p before it is rewritten
    }

    // ---- epilogue: C layout = lanes 0-15 cols, VGPR v -> row v (+8 for hi half) ----
#pragma unroll
    for (int tm = 0; tm < 2; ++tm) {
        const int row0 = rowBase + wm * 32 + tm * 16 + half * 8;
#pragma unroll
        for (int tn = 0; tn < 4; ++tn) {
            const int col = colBase + wn * 64 + tn * 16 + l15;
            float* o = out + (size_t)row0 * N + col;
#pragma unroll
            for (int v = 0; v < 8; ++v)
                o[(size_t)v * N] = acc[tm][tn][v];
        }
    }
}

extern "C" void kernel_launch(void* const* d_in, const int* in_sizes, int n_in,
                              void* d_out, int out_size, void* d_ws, size_t ws_size,
                              hipStream_t stream) {
    const float* inputs   = (const float*)d_in[0];
    const float* relation = (const float*)d_in[1];
    float* out            = (float*)d_out;

    const int D = in_sizes[1];           // 512
    const int N = in_sizes[0] / D;       // 8192
    (void)n_in; (void)out_size; (void)d_ws; (void)ws_size;

    dim3 grid(N / TILE, N / TILE);       // 64 x 64 workgroups
    dim3 block(256);                     // 8 waves
    distmult_wmma_f32_kernel<<<grid, block, 0, stream>>>(inputs, relation, out, N);
}